// NoisyQuantumLayer_11716670783550
// MI455X (gfx1250) — compile-verified
//
#include <hip/hip_runtime.h>
#include <math.h>

typedef __attribute__((ext_vector_type(2))) float v2f;
typedef __attribute__((ext_vector_type(4))) float v4f;
typedef __attribute__((ext_vector_type(8))) float v8f;

#define NQ 4
#define DIM 256
#define ROWS_PER_BLOCK 128
#define THREADS 256

// ============================================================================
// Setup kernel: Heisenberg back-propagation of Z_j observables through the
// fixed part of the channel, in the 4-qubit Pauli basis (256 strings).
// Pauli digit of string p for qubit i: (p >> (2*(3-i))) & 3  (0=I,1=X,2=Y,3=Z)
// Output layout for the main kernel: Tout[rest*16 + j*4 + d0],
//   where p = d0*64 + rest (d0 = qubit-0 digit).
// ============================================================================

// Heisenberg rotation: t'[a] = c*t[a] + s*t[b]; t'[b] = c*t[b] - s*t[a]
//   RZ_H: (a,b)=(X,Y)=(1,2)   RX_H: (a,b)=(Y,Z)=(2,3)   RY_H: (a,b)=(Z,X)=(3,1)
__device__ __forceinline__ void rot_pair_h(float* t, int idx, int p, int qubit,
                                           int a, int b, float c, float s) {
  const int sh = 2 * (3 - qubit);
  const int d = (p >> sh) & 3;
  float self = t[idx];
  float nv = self;
  if (d == a) {
    nv = c * self + s * t[(idx & ~(3 << sh)) | (b << sh)];
  } else if (d == b) {
    nv = c * self - s * t[(idx & ~(3 << sh)) | (a << sh)];
  }
  __syncthreads();
  t[idx] = nv;
  __syncthreads();
}

// CNOT conjugation (self-adjoint, involutive signed permutation of digit pairs)
__device__ __forceinline__ void cnot_h(float* t, int idx, int p, int cq, int tq) {
  const int shc = 2 * (3 - cq), sht = 2 * (3 - tq);
  const int dc = (p >> shc) & 3, dt = (p >> sht) & 3;
  const int pr = dc * 4 + dt;
  // II IX IY IZ | XI XX XY XZ | YI YX YY YZ | ZI ZX ZY ZZ
  const int   map[16] = {0, 1, 14, 15, 5, 4, 11, 10, 9, 8, 7, 6, 12, 13, 2, 3};
  const float sgn[16] = {1.f,1.f,1.f,1.f, 1.f,1.f,1.f,-1.f,
                         1.f,1.f,-1.f,1.f, 1.f,1.f,1.f,1.f};
  const int np = map[pr];
  int q = idx & ~((3 << shc) | (3 << sht));
  q |= ((np >> 2) << shc) | ((np & 3) << sht);
  float val = sgn[pr] * t[q];
  __syncthreads();
  t[idx] = val;
  __syncthreads();
}

__global__ __launch_bounds__(1024) void qptm_setup(const float* __restrict__ qw,
                                                   float* __restrict__ Tout) {
  __shared__ float t[4 * 256];
  const int tid = threadIdx.x;
  const int p = tid & 255, j = tid >> 8;
  // observable Z_j has coefficient 1 on the string with Z on qubit j
  t[tid] = (p == (3 << (2 * (3 - j)))) ? 1.0f : 0.0f;
  __syncthreads();

  // depolarize (self-adjoint): scale by k^(#non-identity digits), k = 1-4p/3
  const float kdep = 1.0f - 4.0f * 0.05f / 3.0f;
  float dep = 1.0f;
#pragma unroll
  for (int q = 0; q < 4; ++q)
    if (((p >> (2 * q)) & 3) != 0) dep *= kdep;

  // --- reverse of: [layer1: CNOT,RY,RZ per qubit] ; depolarize ---
  t[tid] *= dep;
  __syncthreads();
  for (int i = 3; i >= 0; --i) {
    float c1 = __cosf(qw[4 + i]), s1 = __sinf(qw[4 + i]);
    rot_pair_h(t, tid, p, i, 1, 2, c1, s1);  // RZ adjoint
    rot_pair_h(t, tid, p, i, 3, 1, c1, s1);  // RY adjoint
    cnot_h(t, tid, p, i, (i + 1) & 3);
  }
  // --- reverse of: [fixed RX,RZ per qubit] ; depolarize ---
  t[tid] *= dep;
  __syncthreads();
  for (int i = 3; i >= 0; --i) {
    float c0 = __cosf(qw[i]), s0 = __sinf(qw[i]);
    rot_pair_h(t, tid, p, i, 1, 2, c0, s0);  // RZ adjoint
    rot_pair_h(t, tid, p, i, 2, 3, c0, s0);  // RX adjoint
  }

  const int d0 = p >> 6, rest = p & 63;
  Tout[rest * 16 + j * 4 + d0] = t[tid];
}

// branch-free tanh: (e-1)/(e+1), e = exp(2v)   -> v_exp_f32 + v_rcp_f32
__device__ __forceinline__ float fast_tanh(float v) {
  float e = __expf(2.0f * v);
  return __fdividef(e - 1.0f, e + 1.0f);
}

// ============================================================================
// Main kernel: per wave, 16 rows.
//   stage 1: pre = x @ W_in^T  via 64x V_WMMA_F32_16X16X4_F32
//            (B operand zero-padded to 16 columns in LDS -> no EXEC masking)
//   stage 2: tanh + LayerNorm + Bloch vectors + contraction with T (lanes 0-15)
//   stage 3: out = x + b_out + z @ W_out^T via 16x V_WMMA_F32_16X16X4_F32
// x re-read in stage 3 hits L2 (64MB << 192MB), so HBM traffic ~ 128MB total.
// ============================================================================
__global__ __launch_bounds__(THREADS) void qmain(
    const float* __restrict__ x, const float* __restrict__ Win,
    const float* __restrict__ bin, const float* __restrict__ gamma,
    const float* __restrict__ beta, const float* __restrict__ Wout,
    const float* __restrict__ bout, const float* __restrict__ Tg,
    float* __restrict__ out) {
  __shared__ float Tl[1024];      // T, layout [rest][j][d0]
  __shared__ float WinL[16*256];  // W_in zero-padded to 16 rows [16][256]
  __shared__ float WoutL[1024];   // W_out [256][4]
  __shared__ float boL[256];      // b_out
  __shared__ float xpL[ROWS_PER_BLOCK * 4];
  __shared__ float zL[ROWS_PER_BLOCK * 4];

  const int tid = threadIdx.x;
  for (int i = tid; i < 1024; i += THREADS) {
    Tl[i] = Tg[i];
    WoutL[i] = Wout[i];
  }
  for (int i = tid; i < 16 * 256; i += THREADS)
    WinL[i] = (i < NQ * 256) ? Win[i] : 0.0f;
  boL[tid] = bout[tid];
  __syncthreads();

  const int wave = tid >> 5, lane = tid & 31;
  const int half = lane >> 4, l16 = lane & 15;
  const int rowBase = blockIdx.x * ROWS_PER_BLOCK + wave * 16;

  // ---------------- stage 1: x @ W_in^T (M=16, N=16 padded, K=256) ---------
  v8f acc = {0.f, 0.f, 0.f, 0.f, 0.f, 0.f, 0.f, 0.f};
  const float* xrow = x + (size_t)(rowBase + l16) * DIM + 2 * half;
  const float* wrow = &WinL[l16 * DIM + 2 * half];
#pragma unroll 8
  for (int k0 = 0; k0 < DIM; k0 += 4) {
    // A 16x4 f32: vgpr0 = K=(0|2), vgpr1 = K=(1|3) per lane half
    v2f a = *(const v2f*)(xrow + k0);
    v2f b = *(const v2f*)(wrow + k0);  // rows >= 4 are zero padding
    acc = __builtin_amdgcn_wmma_f32_16x16x4_f32(false, a, false, b, (short)0,
                                                acc, false, false);
  }
  // C layout: vgpr v, lane -> row (v + 8*half), col l16. Only cols 0..3 valid.
  if (l16 < NQ) {
    const float bi = bin[l16];
#pragma unroll
    for (int v = 0; v < 8; ++v)
      xpL[(wave * 16 + v + 8 * half) * 4 + l16] = fast_tanh(acc[v] + bi);
  }
  // wave-local LDS ordering: no block barrier needed (in-order DS per wave,
  // each wave only touches its own 16-row slice of xpL/zL).

  // ---------------- stage 2: LN + Pauli-transfer contraction ---------------
  if (half == 0) {
    v4f xp = *(const v4f*)(&xpL[(wave * 16 + l16) * 4]);
    float mu = 0.25f * (xp.x + xp.y + xp.z + xp.w);
    float d0 = xp.x - mu, d1 = xp.y - mu, d2 = xp.z - mu, d3 = xp.w - mu;
    float var = 0.25f * (d0 * d0 + d1 * d1 + d2 * d2 + d3 * d3);
    float rs = rsqrtf(var + 1e-5f);
    float a0 = d0 * rs * gamma[0] + beta[0];
    float a1 = d1 * rs * gamma[1] + beta[1];
    float a2 = d2 * rs * gamma[2] + beta[2];
    float a3 = d3 * rs * gamma[3] + beta[3];

    // Bloch coeffs of RZ(t)RX(t)|0>: (1, sin^2 t, -sin t cos t, cos t)
    float s0[4], s1[4], s2[4], s3[4];
    {
      float sn = __sinf(a0), cs = __cosf(a0);
      s0[0] = 1.f; s0[1] = sn * sn; s0[2] = -sn * cs; s0[3] = cs;
    }
    {
      float sn = __sinf(a1), cs = __cosf(a1);
      s1[0] = 1.f; s1[1] = sn * sn; s1[2] = -sn * cs; s1[3] = cs;
    }
    {
      float sn = __sinf(a2), cs = __cosf(a2);
      s2[0] = 1.f; s2[1] = sn * sn; s2[2] = -sn * cs; s2[3] = cs;
    }
    {
      float sn = __sinf(a3), cs = __cosf(a3);
      s3[0] = 1.f; s3[1] = sn * sn; s3[2] = -sn * cs; s3[3] = cs;
    }

    float z0 = 0.f, z1 = 0.f, z2 = 0.f, z3 = 0.f;
#pragma unroll
    for (int r1 = 0; r1 < 4; ++r1) {
      float f1 = s1[r1];
#pragma unroll
      for (int r2 = 0; r2 < 4; ++r2) {
        float f12 = f1 * s2[r2];
#pragma unroll
        for (int r3 = 0; r3 < 4; ++r3) {
          float f = f12 * s3[r3];
          const float* tp = &Tl[(r1 * 16 + r2 * 4 + r3) << 4];
          v4f q0 = *(const v4f*)(tp + 0);   // wave-uniform LDS broadcast
          v4f q1 = *(const v4f*)(tp + 4);
          v4f q2 = *(const v4f*)(tp + 8);
          v4f q3 = *(const v4f*)(tp + 12);
          z0 += f * (q0.x * s0[0] + q0.y * s0[1] + q0.z * s0[2] + q0.w * s0[3]);
          z1 += f * (q1.x * s0[0] + q1.y * s0[1] + q1.z * s0[2] + q1.w * s0[3]);
          z2 += f * (q2.x * s0[0] + q2.y * s0[1] + q2.z * s0[2] + q2.w * s0[3]);
          z3 += f * (q3.x * s0[0] + q3.y * s0[1] + q3.z * s0[2] + q3.w * s0[3]);
        }
      }
    }
    v4f zv = {z0, z1, z2, z3};
    *(v4f*)(&zL[(wave * 16 + l16) * 4]) = zv;
  }

  // ---------------- stage 3: out = x + b_out + z @ W_out^T -----------------
  // A = z tile 16x4 (one WMMA A operand), 16 chunks of N=16 over D=256.
  v2f az = *(const v2f*)(&zL[(wave * 16 + l16) * 4 + 2 * half]);
#pragma unroll 2
  for (int n0 = 0; n0 < DIM; n0 += 16) {
    v2f bw = *(const v2f*)(&WoutL[(n0 + l16) * 4 + 2 * half]);
    float cb = boL[n0 + l16];
    v8f cc;
#pragma unroll
    for (int v = 0; v < 8; ++v)
      cc[v] = x[(size_t)(rowBase + v + 8 * half) * DIM + n0 + l16] + cb;
    v8f dd = __builtin_amdgcn_wmma_f32_16x16x4_f32(false, az, false, bw,
                                                   (short)0, cc, false, false);
#pragma unroll
    for (int v = 0; v < 8; ++v)
      out[(size_t)(rowBase + v + 8 * half) * DIM + n0 + l16] = dd[v];
  }
}

extern "C" void kernel_launch(void* const* d_in, const int* in_sizes, int n_in,
                              void* d_out, int out_size, void* d_ws, size_t ws_size,
                              hipStream_t stream) {
  const float* x     = (const float*)d_in[0];
  const float* Win   = (const float*)d_in[1];
  const float* bin   = (const float*)d_in[2];
  const float* gamma = (const float*)d_in[3];
  const float* beta  = (const float*)d_in[4];
  const float* qw    = (const float*)d_in[5];
  const float* Wout  = (const float*)d_in[6];
  const float* bout  = (const float*)d_in[7];
  float* out = (float*)d_out;
  float* T = (float*)d_ws;  // 1024 floats

  const int rows = in_sizes[0] / DIM;
  qptm_setup<<<1, 1024, 0, stream>>>(qw, T);
  qmain<<<rows / ROWS_PER_BLOCK, THREADS, 0, stream>>>(x, Win, bin, gamma, beta,
                                                       Wout, bout, T, out);
}